// CustomEmbeddingCollection_13761075216722
// MI455X (gfx1250) — compile-verified
//
#include <hip/hip_runtime.h>
#include <stdint.h>

// out[0:T]  = table_a[indices_a]   (T rows of 64 f32 = 256 B/row)
// out[T:2T] = table_b[indices_b]
// Pure gather: bandwidth-bound (~800 MB total traffic -> ~34us @ 23.3 TB/s).
// Path: gfx1250 async DMA  global -> LDS -> global  (ASYNCcnt-tracked,
// per-lane b128, data never occupies VGPRs). Phase-split so the 8 index
// loads clause behind a single s_wait_loadcnt, then 8 async loads in
// flight per lane before one s_wait_asynccnt; NT hint on output stores.

typedef float f32x4 __attribute__((ext_vector_type(4)));
typedef int   v4i   __attribute__((vector_size(16)));   // builtin param type

#define AS_GLOBAL __attribute__((address_space(1)))
#define AS_LDS    __attribute__((address_space(3)))

#if defined(__has_builtin)
#if __has_builtin(__builtin_amdgcn_global_load_async_to_lds_b128) && \
    __has_builtin(__builtin_amdgcn_global_store_async_from_lds_b128)
#define HAVE_ASYNC_LDS 1
#endif
#endif

__device__ __forceinline__ void wait_asynccnt0() {
  asm volatile("s_wait_asynccnt 0" ::: "memory");
}

#ifdef HAVE_ASYNC_LDS

// ---------------- async global->LDS->global DMA gather ----------------
// One launch per table. chunk = 16B piece of the output; 16 chunks per row.
template <int UNROLL>
__global__ __launch_bounds__(256) void gather_async_kernel(
    const int* __restrict__ idx, const f32x4* __restrict__ tab,
    f32x4* __restrict__ out, int nchunks)
{
  __shared__ f32x4 lbuf[256 * UNROLL];                 // 16 B per lane per step
  const int tid  = threadIdx.x;
  const int blk0 = blockIdx.x * (256 * UNROLL);
  const int base = blk0 + tid;

  if (blk0 + 256 * UNROLL <= nchunks) {
    // ---------- fast path: full block, no per-step guards ----------
    // Phase 1: gather the 8 row indices (clause of b32 loads, one wait).
    unsigned e[UNROLL];
#pragma unroll
    for (int k = 0; k < UNROLL; ++k)
      e[k] = (unsigned)idx[(base + k * 256) >> 4];     // 1 addr per 16 lanes

    // Phase 2: 8 async DMA loads global -> LDS, all in flight.
    const int c = base & 15;                           // float4 within row
#pragma unroll
    for (int k = 0; k < UNROLL; ++k) {
      const f32x4* src = &tab[(long long)e[k] * 16 + c];
      __builtin_amdgcn_global_load_async_to_lds_b128(
          (AS_GLOBAL v4i*)src,
          (AS_LDS v4i*)&lbuf[k * 256 + tid],
          /*offset=*/0, /*cpol=*/0);
    }
    wait_asynccnt0();                                  // all loads landed

    // Phase 3: 8 async DMA stores LDS -> global, NT (write-once stream).
#pragma unroll
    for (int k = 0; k < UNROLL; ++k) {
      __builtin_amdgcn_global_store_async_from_lds_b128(
          (AS_GLOBAL v4i*)&out[base + k * 256],
          (AS_LDS v4i*)&lbuf[k * 256 + tid],
          /*offset=*/0, /*cpol=*/1);                   // TH_STORE_NT
    }
    wait_asynccnt0();                                  // (S_ENDPGM also waits)
  } else {
    // ---------- tail path: per-step bounds checks ----------
#pragma unroll
    for (int k = 0; k < UNROLL; ++k) {
      int chunk = base + k * 256;
      if (chunk < nchunks) {
        unsigned e = (unsigned)idx[chunk >> 4];
        const f32x4* src = &tab[(long long)e * 16 + (chunk & 15)];
        __builtin_amdgcn_global_load_async_to_lds_b128(
            (AS_GLOBAL v4i*)src,
            (AS_LDS v4i*)&lbuf[k * 256 + tid], 0, 0);
      }
    }
    wait_asynccnt0();
#pragma unroll
    for (int k = 0; k < UNROLL; ++k) {
      int chunk = base + k * 256;
      if (chunk < nchunks) {
        __builtin_amdgcn_global_store_async_from_lds_b128(
            (AS_GLOBAL v4i*)&out[chunk],
            (AS_LDS v4i*)&lbuf[k * 256 + tid], 0, 1);
      }
    }
    wait_asynccnt0();
  }
}

#else  // !HAVE_ASYNC_LDS ------------------------------------------------

__global__ __launch_bounds__(256) void gather_plain_kernel(
    const int* __restrict__ idx, const f32x4* __restrict__ tab,
    f32x4* __restrict__ out, int nchunks)
{
  int i = blockIdx.x * blockDim.x + threadIdx.x;
  if (i < nchunks) {
    unsigned e = (unsigned)idx[i >> 4];
    f32x4 v = tab[(long long)e * 16 + (i & 15)];
    __builtin_nontemporal_store(v, &out[i]);
  }
}

#endif

extern "C" void kernel_launch(void* const* d_in, const int* in_sizes, int n_in,
                              void* d_out, int out_size, void* d_ws, size_t ws_size,
                              hipStream_t stream) {
  (void)n_in; (void)d_ws; (void)ws_size; (void)out_size;
  const int*   idx_a = (const int*)d_in[0];
  const int*   idx_b = (const int*)d_in[1];
  const f32x4* tab_a = (const f32x4*)d_in[2];
  const f32x4* tab_b = (const f32x4*)d_in[3];
  f32x4*       out   = (f32x4*)d_out;

  const long long T  = (long long)in_sizes[0];         // 819200
  const int nchunks  = (int)(T * 16);                  // 13,107,200 per table
  f32x4* out_b       = out + T * 16;                   // second half of output

#ifdef HAVE_ASYNC_LDS
  constexpr int UNROLL = 8;
  const int perblk = 256 * UNROLL;                     // 2048 chunks / block
  const int grid   = (nchunks + perblk - 1) / perblk;  // 6400 blocks / table
  gather_async_kernel<UNROLL><<<grid, 256, 0, stream>>>(idx_a, tab_a, out,   nchunks);
  gather_async_kernel<UNROLL><<<grid, 256, 0, stream>>>(idx_b, tab_b, out_b, nchunks);
#else
  const int grid = (nchunks + 255) / 256;
  gather_plain_kernel<<<grid, 256, 0, stream>>>(idx_a, tab_a, out,   nchunks);
  gather_plain_kernel<<<grid, 256, 0, stream>>>(idx_b, tab_b, out_b, nchunks);
#endif
}